// DeTPPLoss_35098472743335
// MI455X (gfx1250) — compile-verified
//
#include <hip/hip_runtime.h>
#include <math.h>

// Problem constants (match reference)
static constexpr int cB = 64;
static constexpr int cK = 4;
static constexpr int cC = 256;
static constexpr int cN = 256 * 64;     // I*B = 16384
static constexpr int GROUPS = cN / 16;  // 1024 groups of 16 n's

typedef __attribute__((ext_vector_type(2))) float v2f;
typedef __attribute__((ext_vector_type(8))) float v8f;

// itertools.permutations(range(4)) lexicographic order (compile-time only)
constexpr unsigned char PERMS_C[24][4] = {
    {0,1,2,3},{0,1,3,2},{0,2,1,3},{0,2,3,1},{0,3,1,2},{0,3,2,1},
    {1,0,2,3},{1,0,3,2},{1,2,0,3},{1,2,3,0},{1,3,0,2},{1,3,2,0},
    {2,0,1,3},{2,0,3,1},{2,1,0,3},{2,1,3,0},{2,3,0,1},{2,3,1,0},
    {3,0,1,2},{3,0,2,1},{3,1,0,2},{3,1,2,0},{3,2,0,1},{3,2,1,0}};

// Bitmask over perms p of S[p][kt] = (perm[p][kt>>2] == (kt&3)); bits 24..31 == 0.
constexpr unsigned smask(int kt) {
    unsigned m = 0;
    for (int p = 0; p < 24; ++p)
        if (PERMS_C[p][kt >> 2] == (unsigned char)(kt & 3)) m |= (1u << p);
    return m;
}
// Spot-check against hand-computed values.
static_assert(smask(0)  == 0x0000003Fu, "smask0");
static_assert(smask(5)  == 0x0030C003u, "smask5");
static_assert(smask(9)  == 0x00861014u, "smask9");
static_assert(smask(14) == 0x00140492u, "smask14");

constexpr unsigned SM[16] = {
    smask(0),  smask(1),  smask(2),  smask(3),
    smask(4),  smask(5),  smask(6),  smask(7),
    smask(8),  smask(9),  smask(10), smask(11),
    smask(12), smask(13), smask(14), smask(15)};

__device__ __forceinline__ float sel4(float a0, float a1, float a2, float a3, int s) {
    return (s == 0) ? a0 : (s == 1) ? a1 : (s == 2) ? a2 : a3;
}
__device__ __forceinline__ int sel4i(int a0, int a1, int a2, int a3, int s) {
    return (s == 0) ? a0 : (s == 1) ? a1 : (s == 2) ? a2 : a3;
}

// Per group of 16 n's: build 16x16 cost matrix (rows kt=k*4+t, cols n) in LDS,
// then Totals(24x16) = S(24x16) x Cost(16x16) via v_wmma_f32_16x16x4_f32, min over perms.
__global__ __launch_bounds__(256) void detpp_cost_kernel(
    const float* __restrict__ g_time, const float* __restrict__ g_amount,
    const float* __restrict__ g_out_time, const float* __restrict__ g_out_amount,
    const float* __restrict__ g_out_logits, const int* __restrict__ g_cat,
    const int* __restrict__ g_lengths, const int* __restrict__ g_indices,
    float* __restrict__ g_partials, float* __restrict__ g_counts)
{
    __shared__ float cost_lds[16][17];  // [n-in-group][kt], stride 17 -> bank-conflict-free
    __shared__ float valid_lds[16];

    const int g    = blockIdx.x;        // group of 16 n's
    const int wave = threadIdx.x >> 5;  // 0..7
    const int lane = threadIdx.x & 31;

    // ---------------- Phase 1: each wave computes cost for 2 n's ----------------
    for (int s = 0; s < 2; ++s) {
        const int j = wave * 2 + s;     // 0..15 within group
        const int n = g * 16 + j;       // flat n = i*B + b (matches indices flat layout)
        const int b = n & (cB - 1);
        const int idx = g_indices[n];
        const float validf = ((idx + cK) < g_lengths[b]) ? 1.0f : 0.0f;

        const size_t base_lb = (size_t)idx * cB + b;              // (idx, b) in (L,B)
        const float* logits = g_out_logits + base_lb * (cK * cC); // 1024 contiguous floats

        // log-sum-exp per k over C=256 (wave-cooperative, 8 floats/lane)
        float lse[4];
        #pragma unroll
        for (int k = 0; k < 4; ++k) {
            const float* row = logits + k * cC;
            float4 u0 = *(const float4*)(row + lane * 4);        // floats   0..127
            float4 u1 = *(const float4*)(row + 128 + lane * 4);  // floats 128..255
            float mx = fmaxf(fmaxf(fmaxf(u0.x, u0.y), fmaxf(u0.z, u0.w)),
                             fmaxf(fmaxf(u1.x, u1.y), fmaxf(u1.z, u1.w)));
            #pragma unroll
            for (int off = 16; off >= 1; off >>= 1) mx = fmaxf(mx, __shfl_xor(mx, off, 32));
            float sm = __expf(u0.x - mx) + __expf(u0.y - mx) + __expf(u0.z - mx) + __expf(u0.w - mx)
                     + __expf(u1.x - mx) + __expf(u1.y - mx) + __expf(u1.z - mx) + __expf(u1.w - mx);
            #pragma unroll
            for (int off = 16; off >= 1; off >>= 1) sm += __shfl_xor(sm, off, 32);
            lse[k] = mx + __logf(sm);
        }

        // targets (wave-uniform addresses)
        const float t0 = g_time[base_lb];
        int tc[4]; float tdt[4]; float ta1[4];
        #pragma unroll
        for (int t = 0; t < 4; ++t) {
            const size_t lb = base_lb + (size_t)(t + 1) * cB;
            tc[t]  = g_cat[lb];
            tdt[t] = g_time[lb] - t0;
            ta1[t] = g_amount[lb];
        }

        if (lane < 16) {
            const int k = lane >> 2, t = lane & 3;
            const int   tci  = sel4i(tc[0], tc[1], tc[2], tc[3], t);
            const float tdtt = sel4(tdt[0], tdt[1], tdt[2], tdt[3], t);
            const float ta1t = sel4(ta1[0], ta1[1], ta1[2], ta1[3], t);
            const float lsek = sel4(lse[0], lse[1], lse[2], lse[3], k);
            const float lg = logits[k * cC + tci];               // L0/L2 hit (just streamed)
            const float ot = g_out_time[base_lb * cK + k];
            const float oa = g_out_amount[base_lb * cK + k];
            const float ce = lsek - lg;                          // -log_softmax at true cat
            cost_lds[j][lane] = ce + fabsf(ot - tdtt) + fabsf(oa - ta1t);
        }
        if (lane == 0) valid_lds[j] = validf;
    }
    __syncthreads();
    if (wave != 0) return;

    // ---------------- Phase 2 (wave 0): WMMA permutation totals ----------------
    // Totals(32x16) in 2 tiles: tile0 = perms 0..15, tile1 = perms 16..31 (rows 24..31 zero
    // automatically, since SM bits >= 24 are clear).
    // A(16x4 f32): v0 -> K=+0 (lanes 0-15) / K=+2 (lanes 16-31); v1 -> K=+1 / K=+3.
    // B(4x16 f32): same paired-row rule; cols = lane&15.
    const int m  = lane & 15;
    const int hi = lane >> 4;
    v8f acc0 = {0.f,0.f,0.f,0.f,0.f,0.f,0.f,0.f};
    v8f acc1 = {0.f,0.f,0.f,0.f,0.f,0.f,0.f,0.f};
    #pragma unroll
    for (int c = 0; c < 4; ++c) {
        const int r0 = 4 * c + (hi ? 2 : 0);
        const int r1 = 4 * c + (hi ? 3 : 1);
        // Masks become inline literals after unroll; hi-select is one v_cndmask each.
        const unsigned m0 = hi ? SM[4 * c + 2] : SM[4 * c + 0];
        const unsigned m1 = hi ? SM[4 * c + 3] : SM[4 * c + 1];
        v2f bmat;
        bmat.x = cost_lds[m][r0];
        bmat.y = cost_lds[m][r1];
        v2f a0;                       // tile 0: perms m (0..15)
        a0.x = ((m0 >> m) & 1u) ? 1.0f : 0.0f;
        a0.y = ((m1 >> m) & 1u) ? 1.0f : 0.0f;
        acc0 = __builtin_amdgcn_wmma_f32_16x16x4_f32(false, a0, false, bmat,
                                                     (short)0, acc0, false, false);
        v2f a1;                       // tile 1: perms 16+m (bits >=24 give 0 padding)
        a1.x = ((m0 >> (16 + m)) & 1u) ? 1.0f : 0.0f;
        a1.y = ((m1 >> (16 + m)) & 1u) ? 1.0f : 0.0f;
        acc1 = __builtin_amdgcn_wmma_f32_16x16x4_f32(false, a1, false, bmat,
                                                     (short)0, acc1, false, false);
    }
    // Min over perms. acc0: VGPR r = perm r (lanes<16) / perm r+8 (lanes>=16), col = lane&15.
    float pm = fminf(fminf(fminf(acc0[0], acc0[1]), fminf(acc0[2], acc0[3])),
                     fminf(fminf(acc0[4], acc0[5]), fminf(acc0[6], acc0[7])));
    pm = fminf(pm, __shfl_xor(pm, 16, 32));  // perms 0..15, all lanes, col = lane&15
    // acc1 low lanes: perms 16..23 (high lanes hold zero-padded perms 24..31 -> never read)
    float qm = fminf(fminf(fminf(acc1[0], acc1[1]), fminf(acc1[2], acc1[3])),
                     fminf(fminf(acc1[4], acc1[5]), fminf(acc1[6], acc1[7])));
    const float vj = (lane < 16) ? valid_lds[lane] : 0.0f;
    float lsum = (lane < 16) ? fminf(pm, qm) * vj : 0.0f;  // min total == sum_k loss[n,k]
    float csum = vj;
    #pragma unroll
    for (int off = 16; off >= 1; off >>= 1) {
        lsum += __shfl_xor(lsum, off, 32);
        csum += __shfl_xor(csum, off, 32);
    }
    if (lane == 0) { g_partials[g] = lsum; g_counts[g] = csum; }
}

// Deterministic final reduction: fixed-order strided sums + shared-memory tree.
__global__ __launch_bounds__(256) void detpp_reduce_kernel(
    const float* __restrict__ g_partials, const float* __restrict__ g_counts,
    float* __restrict__ g_out)
{
    __shared__ float sl[256];
    __shared__ float sc[256];
    const int t = threadIdx.x;
    float ls = 0.f, cs = 0.f;
    for (int i = t; i < GROUPS; i += 256) { ls += g_partials[i]; cs += g_counts[i]; }
    sl[t] = ls; sc[t] = cs;
    __syncthreads();
    for (int off = 128; off >= 1; off >>= 1) {
        if (t < off) { sl[t] += sl[t + off]; sc[t] += sc[t + off]; }
        __syncthreads();
    }
    if (t == 0) g_out[0] = sl[0] / (sc[0] * (float)cK);
}

extern "C" void kernel_launch(void* const* d_in, const int* in_sizes, int n_in,
                              void* d_out, int out_size, void* d_ws, size_t ws_size,
                              hipStream_t stream)
{
    // setup_inputs dict order: time, amount, out_time, out_amount, out_cat_logits,
    //                          cat, lengths, indices
    const float* time_p   = (const float*)d_in[0];
    const float* amount_p = (const float*)d_in[1];
    const float* out_t_p  = (const float*)d_in[2];
    const float* out_a_p  = (const float*)d_in[3];
    const float* out_l_p  = (const float*)d_in[4];
    const int*   cat_p    = (const int*)d_in[5];
    const int*   len_p    = (const int*)d_in[6];
    const int*   idx_p    = (const int*)d_in[7];

    float* partials = (float*)d_ws;
    float* counts   = partials + GROUPS;

    detpp_cost_kernel<<<GROUPS, 256, 0, stream>>>(
        time_p, amount_p, out_t_p, out_a_p, out_l_p, cat_p, len_p, idx_p,
        partials, counts);
    detpp_reduce_kernel<<<1, 256, 0, stream>>>(partials, counts, (float*)d_out);
}